// CombinedLoss_3298534884131
// MI455X (gfx1250) — compile-verified
//
#include <hip/hip_runtime.h>

typedef __attribute__((ext_vector_type(2))) float v2f;
typedef __attribute__((ext_vector_type(8))) float v8f;

#define BATCH 4
#define NPTS  8192
#define TILES (NPTS / 16)                 // 512 tiles of 16 points
#define NCHUNK 4                          // split M-dim across 4 waves per row-tile
#define TILES_PER_CHUNK (TILES / NCHUNK)  // 128
#define WAVES_PER_DIR (BATCH * TILES * NCHUNK)  // 8192

// Raw mins via inline asm: no canonicalize (inputs are ordinary floats here),
// and v_min3 folds two new WMMA results + accumulator into one VALU op.
__device__ __forceinline__ float min3_raw(float acc, float x, float y) {
  float r;
  asm("v_min3_num_f32 %0, %1, %2, %3" : "=v"(r) : "v"(acc), "v"(x), "v"(y));
  return r;
}
__device__ __forceinline__ float min_raw(float a, float b) {
  float r;
  asm("v_min_num_f32 %0, %1, %2" : "=v"(r) : "v"(a), "v"(b));
  return r;
}

// ---------------------------------------------------------------------------
// Prep: swizzle points into WMMA A/B lane layouts, precompute |p|^2 per
// row-lane, and initialize the per-row min buffer to +inf.
//
// A (16x4 MxK), lane L, row r=L%16:  L<16: (p0,p1)   L>=16: (p2, 1.0)
// B (4x16 KxN), lane L, col c=L%16:  L<16: (-2p0,-2p1) L>=16: (-2p2, |p|^2)
// => A x B = |y|^2 - 2 x.y ; loop-invariant C supplies |x_row|^2.
// ---------------------------------------------------------------------------
__global__ void prep_kernel(const float* __restrict__ pts,   // [B, NPTS, 3]
                            v2f* __restrict__ Aout,          // [B*TILES*32]
                            v2f* __restrict__ Bout,          // [B*TILES*32]
                            float* __restrict__ sqout,       // [B*TILES*32]
                            float* __restrict__ rowmin_init) // [B*NPTS]
{
  int i = blockIdx.x * blockDim.x + threadIdx.x;
  if (i >= BATCH * TILES * 32) return;
  int lane = i & 31;
  int t    = (i >> 5) % TILES;
  int b    = i / (TILES * 32);
  int r    = t * 16 + (lane & 15);

  const float* p = pts + ((size_t)b * NPTS + r) * 3;
  float p0 = p[0], p1 = p[1], p2 = p[2];
  float sq = p0 * p0 + p1 * p1 + p2 * p2;

  v2f a, bb;
  if (lane < 16) { a = (v2f){p0, p1};   bb = (v2f){-2.0f * p0, -2.0f * p1}; }
  else           { a = (v2f){p2, 1.0f}; bb = (v2f){-2.0f * p2, sq};         }
  Aout[i]  = a;
  Bout[i]  = bb;
  sqout[i] = sq;                       // |p_{lane%16}|^2, same in both halves

  if (i < BATCH * NPTS) rowmin_init[i] = __builtin_huge_valf();
}

// ---------------------------------------------------------------------------
// Both chamfer directions in one launch. Each wave owns one
// (direction, batch, 16-row tile, M-chunk). Per 16x16 tile:
// one global_load_b64 + one v_wmma_f32_16x16x4_f32 + 4 v_min3.
// ---------------------------------------------------------------------------
__global__ void __launch_bounds__(256)
chamfer_kernel(const v2f* __restrict__ Ax, const v2f* __restrict__ By,
               const float* __restrict__ sqx, float* __restrict__ rmx,
               const v2f* __restrict__ Ay, const v2f* __restrict__ Bx,
               const float* __restrict__ sqy, float* __restrict__ rmy)
{
  int lane = threadIdx.x & 31;
  // wave id is wave-uniform: pin it to SGPRs so the M-loop is a scalar loop
  int gwid = __builtin_amdgcn_readfirstlane(
      (int)((blockIdx.x * blockDim.x + threadIdx.x) >> 5));

  int dir = gwid >= WAVES_PER_DIR;
  int wid = dir ? gwid - WAVES_PER_DIR : gwid;
  const v2f*   A  = dir ? Ay  : Ax;   // row side
  const v2f*   Bm = dir ? Bx  : By;   // col side
  const float* sq = dir ? sqy : sqx;  // row-side norms
  float*       rowmin = dir ? rmy : rmx;

  int chunk = wid % NCHUNK;
  int rt    = (wid / NCHUNK) % TILES;
  int b     = wid / (NCHUNK * TILES);

  v2f a = A[((size_t)b * TILES + rt) * 32 + lane];

  // Loop-invariant C: cfix[k] = |x_{rt*16 + k + 8*(lane>=16)}|^2
  float sqr  = sq[((size_t)b * TILES + rt) * 32 + lane];  // lane L -> row L%16
  int   srcl = (lane & 16) >> 1;                          // 0 or 8
  v8f cfix;
#pragma unroll
  for (int k = 0; k < 8; ++k) cfix[k] = __shfl(sqr, k + srcl, 32);

  const float inf = __builtin_huge_valf();
  v8f mn = {inf, inf, inf, inf, inf, inf, inf, inf};

  const v2f* bp = Bm + (size_t)b * TILES * 32 + lane;
  int t0 = chunk * TILES_PER_CHUNK;
  for (int t = t0; t < t0 + TILES_PER_CHUNK; t += 4) {
    v2f b0 = bp[(size_t)(t + 0) * 32];
    v2f b1 = bp[(size_t)(t + 1) * 32];
    v2f b2 = bp[(size_t)(t + 2) * 32];
    v2f b3 = bp[(size_t)(t + 3) * 32];
    v8f d0 = __builtin_amdgcn_wmma_f32_16x16x4_f32(
        false, a, false, b0, (short)0, cfix, false, false);
    v8f d1 = __builtin_amdgcn_wmma_f32_16x16x4_f32(
        false, a, false, b1, (short)0, cfix, false, false);
    v8f d2 = __builtin_amdgcn_wmma_f32_16x16x4_f32(
        false, a, false, b2, (short)0, cfix, false, false);
    v8f d3 = __builtin_amdgcn_wmma_f32_16x16x4_f32(
        false, a, false, b3, (short)0, cfix, false, false);
#pragma unroll
    for (int k = 0; k < 8; ++k) mn[k] = min3_raw(mn[k], d0[k], d1[k]);
#pragma unroll
    for (int k = 0; k < 8; ++k) mn[k] = min3_raw(mn[k], d2[k], d3[k]);
  }

  // clamp >=0 (monotone, commutes with min); makes uint-ordered atomicMin valid
#pragma unroll
  for (int k = 0; k < 8; ++k) mn[k] = fmaxf(mn[k], 0.0f);

  // min across the 16 lanes of each half (rows k and k+8 live in lane halves)
#pragma unroll
  for (int m = 8; m >= 1; m >>= 1) {
#pragma unroll
    for (int k = 0; k < 8; ++k)
      mn[k] = min_raw(mn[k], __shfl_xor(mn[k], m, 32));
  }

  unsigned int* rm = (unsigned int*)(rowmin + (size_t)b * NPTS + rt * 16);
  if (lane == 0) {
#pragma unroll
    for (int k = 0; k < 8; ++k) atomicMin(rm + k, __float_as_uint(mn[k]));
  } else if (lane == 16) {
#pragma unroll
    for (int k = 0; k < 8; ++k) atomicMin(rm + 8 + k, __float_as_uint(mn[k]));
  }
}

// ---------------------------------------------------------------------------
// Deterministic finalize: sum both row-min buffers + |densities|, combine.
// ---------------------------------------------------------------------------
__global__ void finalize_kernel(const float* __restrict__ rmx,
                                const float* __restrict__ rmy,
                                const float* __restrict__ dens,
                                float* __restrict__ out)
{
  __shared__ float sxw[8], syw[8], sdw[8];
  int tid = threadIdx.x;
  float sx = 0.f, sy = 0.f, sd = 0.f;
  for (int i = tid; i < BATCH * NPTS; i += blockDim.x) {
    sx += rmx[i];
    sy += rmy[i];
    sd += fabsf(dens[i]);
  }
#pragma unroll
  for (int m = 16; m >= 1; m >>= 1) {
    sx += __shfl_xor(sx, m, 32);
    sy += __shfl_xor(sy, m, 32);
    sd += __shfl_xor(sd, m, 32);
  }
  int w = tid >> 5;
  if ((tid & 31) == 0) { sxw[w] = sx; syw[w] = sy; sdw[w] = sd; }
  __syncthreads();
  if (tid == 0) {
    float tx = 0.f, ty = 0.f, td = 0.f;
    for (int i = 0; i < 8; ++i) { tx += sxw[i]; ty += syw[i]; td += sdw[i]; }
    float inv = 1.0f / (float)(BATCH * NPTS);
    float cham = tx * inv + ty * inv;
    cham = fminf(fmaxf(cham, 0.0f), 1.0e6f);
    out[0] = cham + 0.1f * (td * inv);
  }
}

extern "C" void kernel_launch(void* const* d_in, const int* in_sizes, int n_in,
                              void* d_out, int out_size, void* d_ws, size_t ws_size,
                              hipStream_t stream) {
  const float* pred = (const float*)d_in[0];  // [B, N, 3]
  const float* targ = (const float*)d_in[1];  // [B, M, 3]
  const float* dens = (const float*)d_in[2];  // [B, N]
  float* out = (float*)d_out;

  const size_t slots = (size_t)BATCH * TILES * 32;  // 65536 lane-slots
  char* ws = (char*)d_ws;
  v2f*   Ax  = (v2f*)ws;                 // pred as A-side (rows)
  v2f*   Bx  = Ax + slots;               // pred as B-side (cols)
  v2f*   Ay  = Bx + slots;               // target as A-side
  v2f*   By  = Ay + slots;               // target as B-side
  float* sqx = (float*)(By + slots);     // |pred|^2 per row-lane
  float* sqy = sqx + slots;              // |target|^2 per row-lane
  float* rmx = sqy + slots;              // pred->target row mins [B*N]
  float* rmy = rmx + BATCH * NPTS;       // target->pred row mins [B*M]
  // total ws use: 4*slots*8 + 2*slots*4 + 2*B*NPTS*4 = ~2.9 MB

  int prep_blocks = (int)(slots / 256);
  prep_kernel<<<prep_blocks, 256, 0, stream>>>(pred, Ax, Bx, sqx, rmx);
  prep_kernel<<<prep_blocks, 256, 0, stream>>>(targ, Ay, By, sqy, rmy);

  // Both directions in one grid: 16384 waves, 8 waves/block.
  int blocks = 2 * WAVES_PER_DIR * 32 / 256;   // 2048
  chamfer_kernel<<<blocks, 256, 0, stream>>>(Ax, By, sqx, rmx,
                                             Ay, Bx, sqy, rmy);

  finalize_kernel<<<1, 256, 0, stream>>>(rmx, rmy, dens, out);
}